// MattingLaplacianLoss_42193758716267
// MI455X (gfx1250) — compile-verified
//
#include <hip/hip_runtime.h>

// MattingLaplacianLoss for MI455X (gfx1250, wave32).
//
// Algebraic reduction: with d_s = p_s - mean (sum_s d_s == 0), the per-window
// contribution to (xDx - xLx) is
//   contrib = sum_s ||x_s||^2 - (1/9) * ( sum_c u_c^T M u_c + sum_c w_c^2 )
// with M = inv(cov3x3(content window)), w_c = box3x3(styl_c),
// u_c[d] = box3x3(styl_c*cont_d) - w_c * mean_d, mean_d = box3x3(cont_d)/9.
// => 22 per-pixel product maps, each needing a 3x3 box sum, plus a per-window
// 3x3 symmetric inverse. Loss = sum(contrib) / (512*512*4).
//
// CDNA5 mapping: horizontal 3-tap done as banded 16x16 matmul with chained
// V_WMMA_F32_16X16X4_F32 (A-slabs streamed from LDS via ds_load_2addr_b64,
// B = band matrix built in registers); vertical 3-tap in the C/D layout via
// shfl_xor(16) half-swaps. One wave = one 16x16 tile = 14x14 windows.
//
// Phased epilogue keeps peak live set to ~17 v8f accumulators (<192 VGPRs):
//   content boxes -> inv(cov)+mean, then per-channel cross maps folded into a
//   running quadratic-form accumulator. Avoids the >v255 VGPR-MSB region.

typedef __attribute__((ext_vector_type(2))) float v2f;
typedef __attribute__((ext_vector_type(8))) float v8f;

#define TS      18            // LDS tile row stride (dwords): bank-conflict free, 8B aligned
#define NTD     37            // tiles per dim: ceil(510/14)
#define NT_IMG  (NTD * NTD)
#define NTILES  (NT_IMG * 4)  // 5476 wave-tiles
#define NBLOCKS ((NTILES + 7) / 8)
#define HW2     510
#define IMG     512
#define CHS     (IMG * IMG)

__device__ __forceinline__ float2 ld2(const float* t, int r, int c) {
  return *(const float2*)(t + r * TS + c);   // c even, row stride even -> 8B aligned
}

// Horizontal 3-tap via 4 chained WMMA f32 16x16x4 (Y1 = X * R, R banded),
// then vertical 3-tap in C/D register layout (vgpr k holds row k + 8*(lane>=16)).
template <typename F>
__device__ __forceinline__ v8f box3x3(const v2f* bs, F&& gen) {
  v8f acc = {0.f, 0.f, 0.f, 0.f, 0.f, 0.f, 0.f, 0.f};
#pragma unroll
  for (int j = 0; j < 4; ++j) {
    v2f a = gen(j);
    acc = __builtin_amdgcn_wmma_f32_16x16x4_f32(false, a, false, bs[j],
                                                (short)0, acc, false, false);
  }
  float r8 = __shfl_xor(acc[0], 16, 32);  // row 8 (lo half) / row 0 (hi half, unused)
  float r9 = __shfl_xor(acc[1], 16, 32);  // row 9 (lo half)
  v8f o;
  o[0] = acc[0] + acc[1] + acc[2];
  o[1] = acc[1] + acc[2] + acc[3];
  o[2] = acc[2] + acc[3] + acc[4];
  o[3] = acc[3] + acc[4] + acc[5];
  o[4] = acc[4] + acc[5] + acc[6];
  o[5] = acc[5] + acc[6] + acc[7];
  o[6] = acc[6] + acc[7] + r8;            // rows 14/15 on hi half invalid -> masked later
  o[7] = acc[7] + r8 + r9;
  return o;
}

#define GENP(P, Q) [&](int j) -> v2f {                         \
    int cc = 2 * half + 4 * j;                                 \
    float2 x = ld2(P, arow, cc);                               \
    float2 y = ld2(Q, arow, cc);                               \
    v2f a; a.x = x.x * y.x; a.y = x.y * y.y; return a; }
#define GEN1(P) [&](int j) -> v2f {                            \
    int cc = 2 * half + 4 * j;                                 \
    float2 x = ld2(P, arow, cc);                               \
    v2f a; a.x = x.x; a.y = x.y; return a; }

__global__ __launch_bounds__(256) void matting_tiles(
    const float* __restrict__ content,
    const float* __restrict__ stylized,
    float* __restrict__ partial) {
  __shared__ float lds[8][6][16 * TS];   // per-wave: c0,c1,c2,s0,s1,s2 tiles (55.3 KB/block)
  __shared__ float wred[8];

  const int tid  = threadIdx.x;
  const int lane = tid & 31;
  const int wid  = tid >> 5;
  const int half = lane >> 4;   // lane half: selects K pair (A/B) / row offset +8 (C/D)
  const int nIdx = lane & 15;   // N column (B/C/D) == M row (A operand)
  const int arow = nIdx;

  const int wg   = blockIdx.x * 8 + wid;
  const bool live = (wg < NTILES);
  const int w  = live ? wg : 0;            // dead waves still run (uniform barriers, WMMA EXEC)
  const int b  = w / NT_IMG;
  const int t  = w - b * NT_IMG;
  const int tr = t / NTD;
  const int tc = t - tr * NTD;
  const int pr0 = tr * 14, pc0 = tc * 14;

  // ---- stage 6 input tiles (16x16, clamped at image edge) into LDS ----
#pragma unroll
  for (int m = 0; m < 6; ++m) {
    const float* src = (m < 3) ? content  + (b * 3 + m)       * CHS
                               : stylized + (b * 3 + (m - 3)) * CHS;
#pragma unroll
    for (int k = 0; k < 8; ++k) {
      int r  = k + 8 * half;
      int gr = min(pr0 + r,    IMG - 1);
      int gc = min(pc0 + nIdx, IMG - 1);
      lds[wid][m][r * TS + nIdx] = src[gr * IMG + gc];
    }
  }
  __syncthreads();

  // ---- band matrix B slabs for Y = X*R, R[k][c] = (c <= k <= c+2) ----
  v2f bs[4];
#pragma unroll
  for (int j = 0; j < 4; ++j) {
    int K0 = 4 * j + 2 * half;
    bs[j].x = (nIdx <= K0     && K0     <= nIdx + 2) ? 1.f : 0.f;
    bs[j].y = (nIdx <= K0 + 1 && K0 + 1 <= nIdx + 2) ? 1.f : 0.f;
  }

  const float* C0 = lds[wid][0]; const float* C1 = lds[wid][1]; const float* C2 = lds[wid][2];
  const float* S0 = lds[wid][3]; const float* S1 = lds[wid][4]; const float* S2 = lds[wid][5];

  const float i9   = 1.0f / 9.0f;
  const float eps9 = 1e-7f * i9;

  // ---- Phase 1: content boxes -> per-window inv(cov) + mean (content dies) ----
  v8f InvM[6], MeanC[3];
  {
    v8f Scc0 = box3x3(bs, GENP(C0, C0));
    v8f Scc1 = box3x3(bs, GENP(C0, C1));
    v8f Scc2 = box3x3(bs, GENP(C0, C2));
    v8f Scc3 = box3x3(bs, GENP(C1, C1));
    v8f Scc4 = box3x3(bs, GENP(C1, C2));
    v8f Scc5 = box3x3(bs, GENP(C2, C2));
    v8f Tc0  = box3x3(bs, GEN1(C0));
    v8f Tc1  = box3x3(bs, GEN1(C1));
    v8f Tc2  = box3x3(bs, GEN1(C2));
#pragma unroll
    for (int k = 0; k < 8; ++k) {
      float m0 = Tc0[k] * i9, m1 = Tc1[k] * i9, m2 = Tc2[k] * i9;
      float c00 = Scc0[k] * i9 - m0 * m0 + eps9;
      float c01 = Scc1[k] * i9 - m0 * m1;
      float c02 = Scc2[k] * i9 - m0 * m2;
      float c11 = Scc3[k] * i9 - m1 * m1 + eps9;
      float c12 = Scc4[k] * i9 - m1 * m2;
      float c22 = Scc5[k] * i9 - m2 * m2 + eps9;
      float k00 = c11 * c22 - c12 * c12;
      float k01 = c02 * c12 - c01 * c22;
      float k02 = c01 * c12 - c02 * c11;
      float det = c00 * k00 + c01 * k01 + c02 * k02;
      float rd  = 1.0f / det;               // masked windows may be inf/nan -> dropped later
      InvM[0][k] = k00 * rd;
      InvM[1][k] = k01 * rd;
      InvM[2][k] = k02 * rd;
      InvM[3][k] = (c00 * c22 - c02 * c02) * rd;
      InvM[4][k] = (c01 * c02 - c00 * c12) * rd;
      InvM[5][k] = (c00 * c11 - c01 * c01) * rd;
      MeanC[0][k] = m0; MeanC[1][k] = m1; MeanC[2][k] = m2;
    }
  }

  // ---- Phase 2: stylized window sums and squared-norm sum ----
  v8f Wc[3];
  Wc[0] = box3x3(bs, GEN1(S0));
  Wc[1] = box3x3(bs, GEN1(S1));
  Wc[2] = box3x3(bs, GEN1(S2));
  v8f Asq = box3x3(bs, [&](int j) -> v2f {
    int cc = 2 * half + 4 * j;
    float2 x0 = ld2(S0, arow, cc), x1 = ld2(S1, arow, cc), x2 = ld2(S2, arow, cc);
    v2f a;
    a.x = x0.x * x0.x + x1.x * x1.x + x2.x * x2.x;
    a.y = x0.y * x0.y + x1.y * x1.y + x2.y * x2.y;
    return a;
  });

  // ---- Phase 3: per-channel cross maps, folded into quadratic-form acc ----
  v8f qacc = {0.f, 0.f, 0.f, 0.f, 0.f, 0.f, 0.f, 0.f};
#pragma unroll
  for (int c = 0; c < 3; ++c) {
    const float* Sc = (c == 0) ? S0 : (c == 1) ? S1 : S2;
    v8f U0 = box3x3(bs, GENP(Sc, C0));
    v8f U1 = box3x3(bs, GENP(Sc, C1));
    v8f U2 = box3x3(bs, GENP(Sc, C2));
#pragma unroll
    for (int k = 0; k < 8; ++k) {
      float wv = Wc[c][k];
      float u0 = U0[k] - wv * MeanC[0][k];
      float u1 = U1[k] - wv * MeanC[1][k];
      float u2 = U2[k] - wv * MeanC[2][k];
      qacc[k] += InvM[0][k] * u0 * u0 + InvM[3][k] * u1 * u1 + InvM[5][k] * u2 * u2 +
                 2.f * (InvM[1][k] * u0 * u1 + InvM[2][k] * u0 * u2 + InvM[4][k] * u1 * u2);
    }
  }

  // ---- Phase 4: combine, mask, reduce ----
  float lsum = 0.f;
#pragma unroll
  for (int k = 0; k < 8; ++k) {
    int rl = k + 8 * half;
    bool valid = live && (rl < 14) && (nIdx < 14) &&
                 (pr0 + rl < HW2) && (pc0 + nIdx < HW2);
    float wsq = Wc[0][k] * Wc[0][k] + Wc[1][k] * Wc[1][k] + Wc[2][k] * Wc[2][k];
    float contrib = Asq[k] - (qacc[k] + wsq) * i9;
    lsum += valid ? contrib : 0.f;
  }

  // ---- deterministic block reduction -> per-block partial ----
#pragma unroll
  for (int off = 16; off > 0; off >>= 1)
    lsum += __shfl_xor(lsum, off, 32);
  if (lane == 0) wred[wid] = lsum;
  __syncthreads();
  if (tid == 0) {
    float s = 0.f;
#pragma unroll
    for (int i = 0; i < 8; ++i) s += wred[i];
    partial[blockIdx.x] = s;
  }
}

__global__ __launch_bounds__(256) void final_reduce(const float* __restrict__ partial,
                                                    float* __restrict__ out,
                                                    int n, float scale) {
  __shared__ float sm[256];
  float s = 0.f;
  for (int i = threadIdx.x; i < n; i += 256) s += partial[i];
  sm[threadIdx.x] = s;
  __syncthreads();
  for (int off = 128; off > 0; off >>= 1) {
    if (threadIdx.x < off) sm[threadIdx.x] += sm[threadIdx.x + off];
    __syncthreads();
  }
  if (threadIdx.x == 0) out[0] = sm[0] * scale;
}

extern "C" void kernel_launch(void* const* d_in, const int* in_sizes, int n_in,
                              void* d_out, int out_size, void* d_ws, size_t ws_size,
                              hipStream_t stream) {
  const float* content  = (const float*)d_in[0];
  const float* stylized = (const float*)d_in[1];
  float* out     = (float*)d_out;
  float* partial = (float*)d_ws;   // NBLOCKS floats of scratch (< 3 KB)

  matting_tiles<<<NBLOCKS, 256, 0, stream>>>(content, stylized, partial);
  const float scale = 1.0f / (512.f * 512.f * 4.f);   // 1/(HW*B)
  final_reduce<<<1, 256, 0, stream>>>(partial, out, NBLOCKS, scale);
}